// BiMamba_53085795779268
// MI455X (gfx1250) — compile-verified
//
#include <hip/hip_runtime.h>
#include <math.h>

#define D_MODEL 512
#define D_INNER 1024
#define NPROJ   (2 * D_INNER)
#define D_STATE 16
#define DCONV   4
#define DT_RANK 32
#define BATCH   2
#define SEQ     2048
#define MTOT    (BATCH * SEQ)          // 4096 rows per direction

typedef __attribute__((ext_vector_type(2))) float v2f;
typedef __attribute__((ext_vector_type(8))) float v8f;

// ---------------------------------------------------------------------------
// V_WMMA_F32_16X16X4_F32 : D(16x16 f32) = A(16x4 f32) * B(4x16 f32) + C
// 8-arg form (probe-confirmed arity for _16x16x4_*):
//   (neg_a, A, neg_b, B, c_mod, C, reuse_a, reuse_b)
// ---------------------------------------------------------------------------
static __device__ __forceinline__ v8f wmma4(v2f a, v2f b, v8f c) {
  return __builtin_amdgcn_wmma_f32_16x16x4_f32(false, a, false, b, (short)0, c,
                                               false, false);
}

// One wave computes a 16(M) x 64(N) strip of  C = A * W^T.
//   A logical row r lives at a_base + r*a_stride (a_stride may be negative
//   to realize the time-flip of the reverse Mamba branch for free).
//   W is row-major (N_total, K); we use rows n0..n0+63.
// Fragment layout (ISA 7.12.2, f32):
//   A: lane<16 -> k = {k0, k0+1}, lane>=16 -> k = {k0+2, k0+3}; row = lane&15
//   B: symmetric, col = lane&15
//   C: acc[j][r] = C[r + 8*(lane>>4)][lane&15] of N-tile j
static __device__ __forceinline__ void wave_gemm_16x64(
    const float* __restrict__ a_base, int a_stride,
    const float* __restrict__ W, int ldw, int n0, int K, v8f acc[4]) {
  const int lane = threadIdx.x & 31;
  const int half = lane >> 4;
  const int l15  = lane & 15;
  const float* arow = a_base + (ptrdiff_t)l15 * a_stride;
  const float* w0   = W + (size_t)(n0 + l15) * ldw;
  for (int k0 = 0; k0 < K; k0 += 4) {
    const int ka = k0 + 2 * half;
    v2f a;
    a.x = arow[ka];
    a.y = arow[ka + 1];
#pragma unroll
    for (int j = 0; j < 4; ++j) {
      const float* wr = w0 + (size_t)j * 16 * ldw;
      v2f b;
      b.x = wr[ka];
      b.y = wr[ka + 1];
      acc[j] = wmma4(a, b, acc[j]);   // v_wmma_f32_16x16x4_f32
    }
  }
}

static __device__ __forceinline__ float silu_f(float x) {
  return x / (1.0f + __expf(-x));
}
static __device__ __forceinline__ float softplus_f(float x) {
  // stable: logaddexp(x, 0)
  return fmaxf(x, 0.0f) + log1pf(__expf(-fabsf(x)));
}

// ---------------------------------------------------------------------------
// K1: xz = x @ in_proj^T   (reverse dir reads x with negative row stride)
//     cols [0,1024) -> u_pre, cols [1024,2048) -> z
// grid (8, 256, 2)  block 128 (4 waves, each one 16x64 strip)
// ---------------------------------------------------------------------------
__global__ void k_inproj(const float* __restrict__ x,
                         const float* __restrict__ Wf,
                         const float* __restrict__ Wr,
                         float* __restrict__ u_pre,   // [2][MTOT][D_INNER]
                         float* __restrict__ z) {     // [2][MTOT][D_INNER]
  const int wave = threadIdx.x >> 5;
  const int dir  = blockIdx.z;
  const int m0   = blockIdx.y * 16;
  const int n0   = (blockIdx.x * 4 + wave) * 64;
  const int b    = m0 / SEQ;
  const int t0   = m0 % SEQ;

  const float* a_base;
  int a_stride;
  if (dir == 0) {
    a_base   = x + (size_t)m0 * D_MODEL;
    a_stride = D_MODEL;
  } else {
    a_base   = x + ((size_t)b * SEQ + (SEQ - 1 - t0)) * D_MODEL;
    a_stride = -D_MODEL;
  }
  const float* W = dir ? Wr : Wf;

  v8f acc[4] = {};
  wave_gemm_16x64(a_base, a_stride, W, D_MODEL, n0, D_MODEL, acc);

  const int lane = threadIdx.x & 31, half = lane >> 4, l15 = lane & 15;
  float* ub = u_pre + (size_t)dir * MTOT * D_INNER;
  float* zb = z + (size_t)dir * MTOT * D_INNER;
#pragma unroll
  for (int j = 0; j < 4; ++j) {
    const int n = n0 + j * 16 + l15;
    float* dst = (n < D_INNER) ? ub : zb;
    const int nc = (n < D_INNER) ? n : (n - D_INNER);
#pragma unroll
    for (int r = 0; r < 8; ++r) {
      const int m = m0 + 8 * half + r;
      dst[(size_t)m * D_INNER + nc] = acc[j][r];
    }
  }
}

// ---------------------------------------------------------------------------
// K2: depthwise causal conv1d (k=4, left pad 3) + bias + SiLU
// ---------------------------------------------------------------------------
__global__ void k_conv_silu(const float* __restrict__ u_in,
                            const float* __restrict__ cwf,
                            const float* __restrict__ cbf,
                            const float* __restrict__ cwr,
                            const float* __restrict__ cbr,
                            float* __restrict__ u_out) {
  const size_t idx = (size_t)blockIdx.x * blockDim.x + threadIdx.x;
  const int c   = (int)(idx % D_INNER);
  const size_t mm = idx / D_INNER;
  const int dir = (int)(mm / MTOT);
  const int m   = (int)(mm % MTOT);
  const int t   = m % SEQ;

  const float* cw = dir ? cwr : cwf;
  const float* cb = dir ? cbr : cbf;
  const float* ui = u_in + (size_t)dir * MTOT * D_INNER;

  float acc = cb[c];
#pragma unroll
  for (int j = 0; j < DCONV; ++j) {
    const int tt = t - (DCONV - 1) + j;
    if (tt >= 0)
      acc += ui[(size_t)(m - (DCONV - 1) + j) * D_INNER + c] * cw[c * DCONV + j];
  }
  u_out[idx] = silu_f(acc);
}

// ---------------------------------------------------------------------------
// K3: x_dbl = u @ x_proj^T  (N = 64 exactly one 16x64 wave strip)
//     cols [0,32) -> dts_lr output region, cols [32,64) -> B/C scratch
// grid (64, 2) block 128
// ---------------------------------------------------------------------------
__global__ void k_xproj(const float* __restrict__ u,
                        const float* __restrict__ Wf,
                        const float* __restrict__ Wr,
                        float* __restrict__ dtlr_out,   // [2][MTOT][32]
                        float* __restrict__ BC) {       // [2][MTOT][32]
  const int wave = threadIdx.x >> 5;
  const int dir  = blockIdx.y;
  const int m0   = (blockIdx.x * 4 + wave) * 16;
  const float* A = u + ((size_t)dir * MTOT + m0) * D_INNER;
  const float* W = dir ? Wr : Wf;

  v8f acc[4] = {};
  wave_gemm_16x64(A, D_INNER, W, D_INNER, 0, D_INNER, acc);

  const int lane = threadIdx.x & 31, half = lane >> 4, l15 = lane & 15;
#pragma unroll
  for (int j = 0; j < 4; ++j) {
    const int n = j * 16 + l15;
#pragma unroll
    for (int r = 0; r < 8; ++r) {
      const int m = m0 + 8 * half + r;
      const float v = acc[j][r];
      if (n < DT_RANK)
        dtlr_out[((size_t)dir * MTOT + m) * DT_RANK + n] = v;
      else
        BC[((size_t)dir * MTOT + m) * 32 + (n - DT_RANK)] = v;
    }
  }
}

// ---------------------------------------------------------------------------
// K4: delta = softplus(dt_lr @ dt_w^T + dt_b)  -> dts output region
// grid (4, 256, 2) block 128
// ---------------------------------------------------------------------------
__global__ void k_delta(const float* __restrict__ dtlr,
                        const float* __restrict__ Wf,
                        const float* __restrict__ Wr,
                        const float* __restrict__ bf,
                        const float* __restrict__ br,
                        float* __restrict__ delta_out) {  // [2][MTOT][D_INNER]
  const int wave = threadIdx.x >> 5;
  const int dir  = blockIdx.z;
  const int m0   = blockIdx.y * 16;
  const int n0   = (blockIdx.x * 4 + wave) * 64;
  const float* A    = dtlr + ((size_t)dir * MTOT + m0) * DT_RANK;
  const float* W    = dir ? Wr : Wf;
  const float* bias = dir ? br : bf;

  v8f acc[4] = {};
  wave_gemm_16x64(A, DT_RANK, W, DT_RANK, n0, DT_RANK, acc);

  const int lane = threadIdx.x & 31, half = lane >> 4, l15 = lane & 15;
#pragma unroll
  for (int j = 0; j < 4; ++j) {
    const int n  = n0 + j * 16 + l15;
    const float bv = bias[n];
#pragma unroll
    for (int r = 0; r < 8; ++r) {
      const int m = m0 + 8 * half + r;
      delta_out[((size_t)dir * MTOT + m) * D_INNER + n] = softplus_f(acc[j][r] + bv);
    }
  }
}

// ---------------------------------------------------------------------------
// K5: selective scan. One lane per channel, 16-state recurrence in registers.
//     y = (scan_y + u*D) * silu(z)
// grid (4, BATCH, 2) block 256
// ---------------------------------------------------------------------------
__global__ void k_scan(const float* __restrict__ delta,  // [2][MTOT][D_INNER]
                       const float* __restrict__ u,
                       const float* __restrict__ z,
                       const float* __restrict__ BC,     // [2][MTOT][32]
                       const float* __restrict__ Alogf,
                       const float* __restrict__ Alogr,
                       const float* __restrict__ Df,
                       const float* __restrict__ Dr,
                       float* __restrict__ y) {
  const int dir = blockIdx.z;
  const int b   = blockIdx.y;
  const int c   = blockIdx.x * 256 + threadIdx.x;

  const float* Alog = dir ? Alogr : Alogf;
  const float Dc    = (dir ? Dr : Df)[c];

  float A[D_STATE], h[D_STATE];
#pragma unroll
  for (int s = 0; s < D_STATE; ++s) {
    A[s] = -__expf(Alog[c * D_STATE + s]);
    h[s] = 0.0f;
  }

  __shared__ float sB[D_STATE];
  __shared__ float sC[D_STATE];

  for (int t = 0; t < SEQ; ++t) {
    const size_t off = (size_t)dir * MTOT + (size_t)b * SEQ + t;
    __syncthreads();
    if (threadIdx.x < 32) {
      const float v = BC[off * 32 + threadIdx.x];
      if (threadIdx.x < D_STATE) sB[threadIdx.x] = v;
      else                       sC[threadIdx.x - D_STATE] = v;
    }
    __syncthreads();

    const float dt = delta[off * D_INNER + c];
    const float uv = u[off * D_INNER + c];
    const float du = dt * uv;
    float yv = 0.0f;
#pragma unroll
    for (int s = 0; s < D_STATE; ++s) {
      h[s] = __expf(dt * A[s]) * h[s] + du * sB[s];
      yv += h[s] * sC[s];
    }
    const float zv = z[off * D_INNER + c];
    y[off * D_INNER + c] = (yv + uv * Dc) * silu_f(zv);
  }
}

// ---------------------------------------------------------------------------
// K6: out = 0.5 * y @ out_proj^T  (second launch adds at flipped rows)
// grid (2, 256) block 128
// ---------------------------------------------------------------------------
__global__ void k_outproj(const float* __restrict__ y,   // [MTOT][D_INNER]
                          const float* __restrict__ W,   // (D_MODEL, D_INNER)
                          float* __restrict__ out,       // [MTOT][D_MODEL]
                          int flip, int accumulate) {
  const int wave = threadIdx.x >> 5;
  const int m0   = blockIdx.y * 16;
  const int n0   = (blockIdx.x * 4 + wave) * 64;
  const float* A = y + (size_t)m0 * D_INNER;

  v8f acc[4] = {};
  wave_gemm_16x64(A, D_INNER, W, D_INNER, n0, D_INNER, acc);

  const int lane = threadIdx.x & 31, half = lane >> 4, l15 = lane & 15;
  const int b = m0 / SEQ, t0 = m0 % SEQ;
#pragma unroll
  for (int j = 0; j < 4; ++j) {
    const int n = n0 + j * 16 + l15;
#pragma unroll
    for (int r = 0; r < 8; ++r) {
      const int t    = t0 + 8 * half + r;
      const int mout = b * SEQ + (flip ? (SEQ - 1 - t) : t);
      const size_t o = (size_t)mout * D_MODEL + n;
      const float v  = 0.5f * acc[j][r];
      out[o] = accumulate ? (out[o] + v) : v;
    }
  }
}

// ---------------------------------------------------------------------------
extern "C" void kernel_launch(void* const* d_in, const int* in_sizes, int n_in,
                              void* d_out, int out_size, void* d_ws,
                              size_t ws_size, hipStream_t stream) {
  const float* x = (const float*)d_in[0];
  // forward params: 1..9, reverse params: 10..18
  const float* f_in_proj  = (const float*)d_in[1];
  const float* f_conv_w   = (const float*)d_in[2];
  const float* f_conv_b   = (const float*)d_in[3];
  const float* f_x_proj   = (const float*)d_in[4];
  const float* f_dt_w     = (const float*)d_in[5];
  const float* f_dt_b     = (const float*)d_in[6];
  const float* f_A_log    = (const float*)d_in[7];
  const float* f_D        = (const float*)d_in[8];
  const float* f_out_proj = (const float*)d_in[9];
  const float* r_in_proj  = (const float*)d_in[10];
  const float* r_conv_w   = (const float*)d_in[11];
  const float* r_conv_b   = (const float*)d_in[12];
  const float* r_x_proj   = (const float*)d_in[13];
  const float* r_dt_w     = (const float*)d_in[14];
  const float* r_dt_b     = (const float*)d_in[15];
  const float* r_A_log    = (const float*)d_in[16];
  const float* r_D        = (const float*)d_in[17];
  const float* r_out_proj = (const float*)d_in[18];

  float* out = (float*)d_out;
  // output regions: out | dts | dts_lr
  const size_t OUT_ELEMS  = (size_t)BATCH * SEQ * D_MODEL;          // 2,097,152
  const size_t DTS_ELEMS  = (size_t)2 * BATCH * SEQ * D_INNER;      // 8,388,608
  float* dts_out  = out + OUT_ELEMS;
  float* dtlr_out = out + OUT_ELEMS + DTS_ELEMS;

  // workspace layout (floats)
  float* ws = (float*)d_ws;
  const size_t SPD = (size_t)2 * MTOT * D_INNER;   // per-tensor [2][MTOT][1024]
  float* u_pre  = ws;                 // SPD
  float* u_post = ws + SPD;           // SPD
  float* zbuf   = ws + 2 * SPD;       // SPD
  float* ybuf   = ws + 3 * SPD;       // SPD
  float* BC     = ws + 4 * SPD;       // 2*MTOT*32

  // K1: in_proj GEMM (both directions; reverse reads x time-flipped)
  k_inproj<<<dim3(NPROJ / 256, MTOT / 16, 2), dim3(128), 0, stream>>>(
      x, f_in_proj, r_in_proj, u_pre, zbuf);

  // K2: causal depthwise conv + SiLU
  {
    const size_t total = (size_t)2 * MTOT * D_INNER;
    k_conv_silu<<<dim3((unsigned)(total / 256)), dim3(256), 0, stream>>>(
        u_pre, f_conv_w, f_conv_b, r_conv_w, r_conv_b, u_post);
  }

  // K3: x_proj GEMM -> dt_lr (output) + B/C (scratch)
  k_xproj<<<dim3(MTOT / 64, 2), dim3(128), 0, stream>>>(
      u_post, f_x_proj, r_x_proj, dtlr_out, BC);

  // K4: delta GEMM + softplus -> dts output region
  k_delta<<<dim3(D_INNER / 256, MTOT / 16, 2), dim3(128), 0, stream>>>(
      dtlr_out, f_dt_w, r_dt_w, f_dt_b, r_dt_b, dts_out);

  // K5: selective scan + gating
  k_scan<<<dim3(D_INNER / 256, BATCH, 2), dim3(256), 0, stream>>>(
      dts_out, u_post, zbuf, BC, f_A_log, r_A_log, f_D, r_D, ybuf);

  // K6: out_proj; forward writes 0.5*res, reverse adds 0.5*res at flipped rows
  k_outproj<<<dim3(D_MODEL / 256, MTOT / 16), dim3(128), 0, stream>>>(
      ybuf, f_out_proj, out, 0, 0);
  k_outproj<<<dim3(D_MODEL / 256, MTOT / 16), dim3(128), 0, stream>>>(
      ybuf + SPD / 2, r_out_proj, out, 1, 1);
}